// Blocks2Matrix_66073776882374
// MI455X (gfx1250) — compile-verified
//
#include <hip/hip_runtime.h>
#include <hip/hip_bf16.h>

typedef __attribute__((ext_vector_type(2))) float v2f;
typedef __attribute__((ext_vector_type(8))) float v8f;

#define NRAD  8
#define NORB  40        // NRAD * 5
#define NMAT  2560      // 64 atoms * 40 orbitals
#define NMU   5

// ---------------------------------------------------------------------------
// Zero the output Hamiltonian with wide b128 stores (419 MB; harness poisons
// d_out once, and scatter-add needs a clean zero base every call).
// ---------------------------------------------------------------------------
__global__ __launch_bounds__(256) void b2m_zero(float4* __restrict__ out, int n4) {
    int i = blockIdx.x * blockDim.x + threadIdx.x;
    int stride = gridDim.x * blockDim.x;
    float4 z = {0.f, 0.f, 0.f, 0.f};
    for (; i < n4; i += stride) out[i] = z;
}

// ---------------------------------------------------------------------------
// One wave (32 lanes) per sample.
//   OUT[pq, ab] = sum_mu V[pq, mu] * CG2[mu, ab]
//     pq = n1*8 + n2 (64 rows), ab = m1*5 + m2 (25 cols, padded to 32), K = 5.
// Computed as 4 M-tiles x 2 N-tiles of V_WMMA_F32_16X16X4_F32, K=5 split into
// a K=4 WMMA plus a zero-padded K=1 WMMA chained through the accumulator.
// sys/i/j are wave-uniform (one sample per wave): readfirstlane them so the
// H base lives in SGPRs and the f32 atomics use SADDR + 32-bit voffset form.
// ---------------------------------------------------------------------------
__global__ __launch_bounds__(256) void b2m_main(const float* __restrict__ values,
                                                const float* __restrict__ cg,
                                                const int*   __restrict__ sys_idx,
                                                const int*   __restrict__ i_idx,
                                                const int*   __restrict__ j_idx,
                                                float*       __restrict__ H,
                                                int S) {
    // CG2[mu][ab]: ab = m1*5+m2 (<25 valid, rest zero).  cg layout: [m1][m2][mu].
    __shared__ float CG2[NMU][32];
    int tid = threadIdx.x;
    if (tid < NMU * 32) {
        int mu = tid >> 5;
        int ab = tid & 31;
        float v = 0.f;
        if (ab < 25) {
            int m1 = ab / 5, m2 = ab - m1 * 5;
            v = cg[(m1 * 5 + m2) * NMU + mu];
        }
        CG2[mu][ab] = v;
    }
    __syncthreads();

    int wave = (blockIdx.x * blockDim.x + tid) >> 5;   // sample index
    if (wave >= S) return;                              // wave-uniform guard

    int lane    = tid & 31;
    int lane_lo = lane & 15;
    int half    = lane >> 4;                            // 0: K0/K1 rows, 1: K2/K3 rows

    const float* V = values + (size_t)wave * (NMU * NRAD * NRAD);  // [mu][pq]

    // Wave-uniform indices -> SGPRs (one sample per wave by construction).
    int sys = __builtin_amdgcn_readfirstlane(sys_idx[wave]);
    int ii  = __builtin_amdgcn_readfirstlane(i_idx[wave]);
    int jj  = __builtin_amdgcn_readfirstlane(j_idx[wave]);
    float* __restrict__ Hs = H + (size_t)sys * ((size_t)NMAT * NMAT);  // scalar base

    unsigned rbase = (unsigned)(ii * NORB);   // global row base (scalar)
    unsigned cbase = (unsigned)(jj * NORB);   // global col base (scalar)

    // B operands (CG2): 4x16 f32, K rows split across lane halves.
    //   VGPR0: lanes 0-15 = row K0, lanes 16-31 = row K2
    //   VGPR1: lanes 0-15 = row K1, lanes 16-31 = row K3
    v2f Bm[2], B2[2];
    #pragma unroll
    for (int nt = 0; nt < 2; ++nt) {
        int n = nt * 16 + lane_lo;
        float b0 = CG2[half ? 2 : 0][n];
        float b1 = CG2[half ? 3 : 1][n];
        Bm[nt] = (v2f){b0, b1};
        // K=1 pad step: only K-row 0 carries mu=4; everything else zero.
        B2[nt] = (v2f){half ? 0.f : CG2[4][n], 0.f};
    }

    #pragma unroll
    for (int mt = 0; mt < 4; ++mt) {
        int pq = mt * 16 + lane_lo;                    // A-matrix row M = lane&15
        // A: 16x4 f32.  lanes 0-15: VGPR0=K0,VGPR1=K1; lanes 16-31: VGPR0=K2,VGPR1=K3
        float a0 = V[(half ? 2 : 0) * 64 + pq];
        float a1 = V[(half ? 3 : 1) * 64 + pq];
        v2f Am = (v2f){a0, a1};
        // mu=4 column in K-slot 0, zeros elsewhere.
        v2f A2 = (v2f){half ? 0.f : V[4 * 64 + pq], 0.f};

        #pragma unroll
        for (int nt = 0; nt < 2; ++nt) {
            v8f acc = {};
            // (neg_a, A, neg_b, B, c_mod, C, reuse_a, reuse_b)
            acc = __builtin_amdgcn_wmma_f32_16x16x4_f32(
                false, A2, false, B2[nt], (short)0, acc, false, false);
            acc = __builtin_amdgcn_wmma_f32_16x16x4_f32(
                false, Am, false, Bm[nt], (short)0, acc, false, false);

            // D layout: VGPR r -> M = mt*16 + half*8 + r, N = nt*16 + lane_lo.
            int ab = nt * 16 + lane_lo;
            if (ab < 25) {
                unsigned m1 = (unsigned)ab / 5u;
                unsigned m2 = (unsigned)ab - m1 * 5u;
                #pragma unroll
                for (int r = 0; r < 8; ++r) {
                    unsigned pqr = (unsigned)(mt * 16 + half * 8 + r);
                    unsigned n1 = pqr >> 3, n2 = pqr & 7u;
                    unsigned row = rbase + n1 * 5u + m1;   // i*40 + n1*5 + m1
                    unsigned col = cbase + n2 * 5u + m2;   // j*40 + n2*5 + m2
                    float v = acc[r];
                    // 32-bit element offsets against the scalar per-system base
                    unsafeAtomicAdd(Hs + (row * (unsigned)NMAT + col), v);  // (i,j)
                    unsafeAtomicAdd(Hs + (col * (unsigned)NMAT + row), v);  // (j,i)
                }
            }
        }
    }
}

extern "C" void kernel_launch(void* const* d_in, const int* in_sizes, int n_in,
                              void* d_out, int out_size, void* d_ws, size_t ws_size,
                              hipStream_t stream) {
    const float* values  = (const float*)d_in[0];
    const float* cg      = (const float*)d_in[1];
    const int*   sys_idx = (const int*)d_in[2];
    const int*   i_idx   = (const int*)d_in[3];
    const int*   j_idx   = (const int*)d_in[4];
    float*       H       = (float*)d_out;

    int S  = in_sizes[2];          // number of pair samples
    int n4 = out_size / 4;         // float4 count (out_size divisible by 4)

    b2m_zero<<<2048, 256, 0, stream>>>((float4*)d_out, n4);

    int blocks = (S + 7) / 8;      // 8 waves (samples) per 256-thread block
    b2m_main<<<blocks, 256, 0, stream>>>(values, cg, sys_idx, i_idx, j_idx, H, S);
}